// EmoHeadMix_85633057948135
// MI455X (gfx1250) — compile-verified
//
#include <hip/hip_runtime.h>
#include <math.h>

typedef __attribute__((ext_vector_type(4))) float v4f;
typedef __attribute__((ext_vector_type(2))) float v2f;
typedef __attribute__((ext_vector_type(8))) float v8f;
typedef __attribute__((ext_vector_type(4))) unsigned int u32x4;
typedef __attribute__((ext_vector_type(8))) int i32x8;
typedef __attribute__((ext_vector_type(4))) int i32x4;

#define NB    32      // batch
#define NT    4096    // enc time
#define NU    1024    // pred time
#define NC    4       // classes
#define DHALF 512     // E == P
#define DFEAT 1024    // E + P
#define NROWS 128     // B * C
#define SCHUNK 128    // s-rows per masked-sum block
#define LN_EPS 1e-5f

// ---------------------------------------------------------------------------
// Tensor Data Mover: async 2D tile load Global -> LDS (CDNA5 TDM, D# per
// CDNA5 ISA). data_size = 4 bytes. Issued by one wave; tracked
// with TENSORcnt. lds_off is the byte offset of the __shared__ destination
// (low 32 bits of the generic pointer == LDS offset).
// ---------------------------------------------------------------------------
#if __has_builtin(__builtin_amdgcn_tensor_load_to_lds)
#define HAVE_TDM 1
__device__ __forceinline__ void tdm_load_2d_f32(const void* gptr, unsigned lds_off,
                                                unsigned dim0, unsigned dim1,
                                                unsigned tile0, unsigned tile1,
                                                unsigned long long stride0) {
    const unsigned long long ga = (unsigned long long)gptr;
    u32x4 g0;
    g0[0] = 0x1u;                                   // count = 1 valid descriptor
    g0[1] = lds_off;                                // lds_addr [63:32]
    g0[2] = (unsigned)(ga & 0xFFFFFFFFu);           // global_addr [95:64]
    g0[3] = ((unsigned)((ga >> 32) & 0x01FFFFFFull))// global_addr [120:96]
            | 0x80000000u;                          // type = 2 ("image") [127:126]
    i32x8 g1;
    g1[0] = (int)(2u << 16);                        // data_size = 2 (4B); wg_mask = 0
    g1[1] = (int)((dim0 & 0xFFFFu) << 16);          // tensor_dim0[15:0] at [63:48]
    g1[2] = (int)(((dim0 >> 16) & 0xFFFFu) | ((dim1 & 0xFFFFu) << 16));
    g1[3] = (int)(((dim1 >> 16) & 0xFFFFu) | ((tile0 & 0xFFFFu) << 16));
    g1[4] = (int)(tile1 & 0xFFFFu);                 // tile_dim1; tile_dim2 = 0
    g1[5] = (int)(unsigned)(stride0 & 0xFFFFFFFFull);
    g1[6] = (int)(unsigned)((stride0 >> 32) & 0xFFFFull);
    g1[7] = 0;                                      // tensor_dim1_stride unused (1 tile row pass)
    const i32x4 z4 = {0, 0, 0, 0};
#if __clang_major__ >= 23
    const i32x8 z8 = {0, 0, 0, 0, 0, 0, 0, 0};
    __builtin_amdgcn_tensor_load_to_lds(g0, g1, z4, z4, z8, 0);
#else
    __builtin_amdgcn_tensor_load_to_lds(g0, g1, z4, z4, 0);
#endif
}
#else
#define HAVE_TDM 0
#endif

// ---------------------------------------------------------------------------
// Per-batch class counts (exact integer LDS atomics -> deterministic).
// ---------------------------------------------------------------------------
__global__ void count_kernel(const int* __restrict__ labels, int S,
                             float* __restrict__ cnt) {
    __shared__ int scnt[NC];
    const int b = blockIdx.x;
    const int t = threadIdx.x;
    if (t < NC) scnt[t] = 0;
    __syncthreads();
    int c0 = 0, c1 = 0, c2 = 0, c3 = 0;
    for (int s = t; s < S; s += blockDim.x) {
        const int l = labels[b * S + s];
        c0 += (l == 0); c1 += (l == 1); c2 += (l == 2); c3 += (l == 3);
    }
    atomicAdd(&scnt[0], c0); atomicAdd(&scnt[1], c1);
    atomicAdd(&scnt[2], c2); atomicAdd(&scnt[3], c3);
    __syncthreads();
    if (t < NC) cnt[b * NC + t] = (float)scnt[t];
}

// ---------------------------------------------------------------------------
// Masked class sums, stage 1: the bandwidth-bound hot loop (~320 MB stream,
// ~14 us roofline at 23.3 TB/s). Block = 128 threads, each owns one float4
// column group (128*4 = 512 = D). Labels for the block's SCHUNK rows are
// DMA'd into LDS by the TDM (wave 0), then broadcast + readfirstlane so the
// class dispatch is pure SALU. Non-temporal loads: data touched exactly once,
// keep it out of L2. Deterministic per-chunk partials (no float atomics).
// partial layout: [chunk][b*4+c][512]
// ---------------------------------------------------------------------------
__global__ void masked_sum_kernel(const float* __restrict__ x,
                                  const int* __restrict__ labels,
                                  float* __restrict__ partial, int S) {
    __shared__ int slbl[SCHUNK];
    const int b  = blockIdx.y;
    const int s0 = blockIdx.x * SCHUNK;
    const int t  = threadIdx.x;          // 0..127
    const int d  = t * 4;

#if HAVE_TDM
    if (t < 32) {   // one wave issues the DMA; TDM ignores EXEC within the wave
        tdm_load_2d_f32(labels + (size_t)b * S + s0, (unsigned)(size_t)&slbl[0],
                        SCHUNK, 1, SCHUNK, 1, SCHUNK);
        __builtin_amdgcn_s_wait_tensorcnt(0);
    }
#else
    slbl[t] = labels[(size_t)b * S + s0 + t];
#endif
    __syncthreads();

    v4f a0 = {0.f,0.f,0.f,0.f}, a1 = a0, a2 = a0, a3 = a0;
    const float* base = x + ((size_t)b * S + s0) * DHALF + d;

    for (int i = 0; i < SCHUNK; ++i) {
        const v4f v = __builtin_nontemporal_load(
            (const v4f*)(base + (size_t)i * DHALF));
        __builtin_prefetch(base + (size_t)(i + 16) * DHALF, 0, 0);
        const int l = __builtin_amdgcn_readfirstlane(slbl[i]);
        if      (l == 0) a0 += v;
        else if (l == 1) a1 += v;
        else if (l == 2) a2 += v;
        else             a3 += v;
    }

    float* o = partial + ((size_t)blockIdx.x * NROWS + (size_t)b * NC) * DHALF + d;
    *(v4f*)(o + 0 * DHALF) = a0;
    *(v4f*)(o + 1 * DHALF) = a1;
    *(v4f*)(o + 2 * DHALF) = a2;
    *(v4f*)(o + 3 * DHALF) = a3;
}

// ---------------------------------------------------------------------------
// Stage 2: reduce chunk partials -> mean -> concat -> LayerNorm -> feat_n.
// One block per (b,c) row. Threads 0..127 own enc half (d = 4t), threads
// 128..255 own pred half.
// ---------------------------------------------------------------------------
__global__ void finalize_ln_kernel(const float* __restrict__ part_enc,
                                   const float* __restrict__ part_pred,
                                   const float* __restrict__ cnt_enc,
                                   const float* __restrict__ cnt_pred,
                                   const float* __restrict__ norm_w,
                                   const float* __restrict__ norm_b,
                                   float* __restrict__ feat_n) {
    __shared__ float ssum[256];
    __shared__ float ssq[256];
    const int r = blockIdx.x;            // 0..127 : b = r>>2, c = r&3
    const int t = threadIdx.x;           // 0..255

    v4f f;
    if (t < 128) {
        const int d = t * 4;
        v4f s = {0.f,0.f,0.f,0.f};
        #pragma unroll
        for (int ch = 0; ch < NT / SCHUNK; ++ch)
            s += *(const v4f*)(part_enc + ((size_t)(ch * NROWS + r)) * DHALF + d);
        f = s * (1.f / fmaxf(cnt_enc[r], 1.f));
    } else {
        const int d = (t - 128) * 4;
        v4f s = {0.f,0.f,0.f,0.f};
        #pragma unroll
        for (int ch = 0; ch < NU / SCHUNK; ++ch)
            s += *(const v4f*)(part_pred + ((size_t)(ch * NROWS + r)) * DHALF + d);
        f = s * (1.f / fmaxf(cnt_pred[r], 1.f));
    }

    ssum[t] = f.x + f.y + f.z + f.w;
    ssq[t]  = f.x*f.x + f.y*f.y + f.z*f.z + f.w*f.w;
    __syncthreads();
    for (int off = 128; off > 0; off >>= 1) {
        if (t < off) { ssum[t] += ssum[t + off]; ssq[t] += ssq[t + off]; }
        __syncthreads();
    }
    const float mean = ssum[0] * (1.f / DFEAT);
    const float var  = ssq[0] * (1.f / DFEAT) - mean * mean;
    const float rstd = rsqrtf(var + LN_EPS);

    const int dfull = t * 4;             // 0..1023
    const v4f w  = *(const v4f*)(norm_w + dfull);
    const v4f bb = *(const v4f*)(norm_b + dfull);
    v4f outv = (f - mean) * rstd * w + bb;
    *(v4f*)(feat_n + (size_t)r * DFEAT + dfull) = outv;
}

// ---------------------------------------------------------------------------
// Head GEMM via V_WMMA_F32_16X16X4_F32 + log-softmax + loss (single block).
// (128 x 1024) @ (1024 x 4->padded 16). 8 waves x 16 rows x 256 K-steps.
// head_w (1024 x 4 f32 tile) staged into LDS by the TDM.
// A frag (16x4 f32): lane L holds row M=L&15; VGPR0 = K(2*hi), VGPR1 = K(2*hi+1).
// B frag (4x16 f32): lane L holds col N=L&15; VGPR0 = K(2*hi), VGPR1 = K(2*hi+1).
// D (16x16 f32): VGPR i -> row M = i + 8*hi, col N = L&15.
// ---------------------------------------------------------------------------
__global__ void head_loss_kernel(const float* __restrict__ feat_n,
                                 const float* __restrict__ head_w,
                                 const float* __restrict__ head_b,
                                 float* __restrict__ out) {
    __shared__ float hw[NC * 1024];      // head_w staged via TDM
    __shared__ float lg[NROWS * NC];     // logits
    __shared__ float red[NROWS];
    const int t = threadIdx.x;

#if HAVE_TDM
    if (t < 32) {
        tdm_load_2d_f32(head_w, (unsigned)(size_t)&hw[0],
                        1024, NC, 1024, NC, 1024ull);
        __builtin_amdgcn_s_wait_tensorcnt(0);
    }
#else
    for (int i = t; i < NC * 1024; i += 256) hw[i] = head_w[i];
#endif
    __syncthreads();

    const int lane = t & 31;
    const int wv   = t >> 5;             // 0..7
    const int hi   = lane >> 4;          // half-wave
    const int n    = lane & 15;          // N (col / class) for B,D; also M for A
    const int r0   = wv * 16;
    const float bmask = (n < NC) ? 1.f : 0.f;
    const float hb    = bmask * head_b[n & 3];

    const float* A    = feat_n + (size_t)(r0 + n) * DFEAT + 2 * hi;
    const float* Brow = hw + (n & 3) * 1024 + 2 * hi;

    v8f acc = {0.f,0.f,0.f,0.f,0.f,0.f,0.f,0.f};
    for (int kb = 0; kb < DFEAT; kb += 4) {
        v2f a;  a.x  = A[kb];             a.y  = A[kb + 1];
        v2f bf; bf.x = bmask * Brow[kb];  bf.y = bmask * Brow[kb + 1];
        acc = __builtin_amdgcn_wmma_f32_16x16x4_f32(
            /*neg_a=*/false, a, /*neg_b=*/false, bf,
            /*c_mod=*/(short)0, acc, /*reuse_a=*/false, /*reuse_b=*/false);
    }

    #pragma unroll
    for (int i = 0; i < 8; ++i) {
        if (n < NC) lg[(r0 + i + 8 * hi) * NC + n] = acc[i] + hb;
    }
    __syncthreads();

    if (t < NROWS) {
        const float l0 = lg[t*NC+0], l1 = lg[t*NC+1], l2 = lg[t*NC+2], l3 = lg[t*NC+3];
        const float mx = fmaxf(fmaxf(l0, l1), fmaxf(l2, l3));
        const float lse = mx + logf(expf(l0-mx) + expf(l1-mx) + expf(l2-mx) + expf(l3-mx));
        const int c = t & 3;             // diagonal class for row (b,c)
        const float sel = ((c == 0) ? l0 : (c == 1) ? l1 : (c == 2) ? l2 : l3) - lse;
        red[t] = sel;
    }
    __syncthreads();
    for (int off = 64; off > 0; off >>= 1) {
        if (t < off) red[t] += red[t + off];
        __syncthreads();
    }
    if (t == 0) out[0] = -red[0] / (float)NROWS;
}

// ---------------------------------------------------------------------------
// Launch
// ---------------------------------------------------------------------------
extern "C" void kernel_launch(void* const* d_in, const int* in_sizes, int n_in,
                              void* d_out, int out_size, void* d_ws, size_t ws_size,
                              hipStream_t stream) {
    const float* enc   = (const float*)d_in[0];
    const float* pred  = (const float*)d_in[1];
    const int*   flab  = (const int*)d_in[2];
    const int*   ftlab = (const int*)d_in[3];
    // d_in[4] = label (tiled arange) -> statically the diagonal; unused.
    const float* nw    = (const float*)d_in[5];
    const float* nb    = (const float*)d_in[6];
    const float* hwp   = (const float*)d_in[7];
    const float* hbp   = (const float*)d_in[8];

    float* ws = (float*)d_ws;
    const size_t N_PART_ENC  = (size_t)(NT / SCHUNK) * NROWS * DHALF; // 2,097,152
    const size_t N_PART_PRED = (size_t)(NU / SCHUNK) * NROWS * DHALF; //   524,288
    float* part_enc  = ws;
    float* part_pred = part_enc + N_PART_ENC;
    float* cnt_enc   = part_pred + N_PART_PRED;
    float* cnt_pred  = cnt_enc + NROWS;
    float* feat_n    = cnt_pred + NROWS;                              // 131,072

    count_kernel<<<NB, 256, 0, stream>>>(flab, NT, cnt_enc);
    count_kernel<<<NB, 256, 0, stream>>>(ftlab, NU, cnt_pred);

    masked_sum_kernel<<<dim3(NT / SCHUNK, NB), 128, 0, stream>>>(enc, flab, part_enc, NT);
    masked_sum_kernel<<<dim3(NU / SCHUNK, NB), 128, 0, stream>>>(pred, ftlab, part_pred, NU);

    finalize_ln_kernel<<<NROWS, 256, 0, stream>>>(part_enc, part_pred, cnt_enc,
                                                  cnt_pred, nw, nb, feat_n);

    head_loss_kernel<<<1, 256, 0, stream>>>(feat_n, hwp, hbp, (float*)d_out);
}